// OccupancyConnectivity_44753559224580
// MI455X (gfx1250) — compile-verified
//
#include <hip/hip_runtime.h>

typedef __attribute__((ext_vector_type(2))) float v2f;
typedef __attribute__((ext_vector_type(8))) float v8f;

#define G 385u
#define PLANE (385u * 385u)          // 148225
#define NTOT (385u * 385u * 385u)    // 57066625 < 2^31
#define NBLOCKS 4096
#define BLOCK 256

// Wave32 reduction of one float per lane via V_WMMA_F32_16X16X4_F32, then
// cross-wave combine through LDS. Valid result returned on threadIdx.x == 0.
//
// Trick (layout-agnostic): A = all-ones 16x4 (64 slots = 2 VGPRs x 32 lanes,
// all set to 1.0 -> all-ones regardless of lane mapping). B = {partial, 0}
// per lane (64 slots hold each lane partial exactly once plus zeros).
// D[m][n] = colsum_n(B), so the sum of ALL 256 elements of D
// (= 8 VGPRs x 32 lanes, each element exactly once, whatever the mapping)
// equals 16 * sum(B) = 16 * wave_total. Divide by 16 (exact).
__device__ __forceinline__ float block_reduce_wmma(float p, float* lds8) {
    v2f a; a.x = 1.0f; a.y = 1.0f;   // all-ones A matrix
    v2f b; b.x = p;    b.y = 0.0f;   // lane partials in B matrix
    v8f c = {};
    // 8 args: (neg_a, A, neg_b, B, c_mod, C, reuse_a, reuse_b)
    v8f d = __builtin_amdgcn_wmma_f32_16x16x4_f32(
        /*neg_a=*/false, a, /*neg_b=*/false, b,
        /*c_mod=*/(short)0, c, /*reuse_a=*/false, /*reuse_b=*/false);
    float s = d[0] + d[1] + d[2] + d[3] + d[4] + d[5] + d[6] + d[7];
    // Cross-lane sum over the wave (wave32): after this every lane holds
    // 16 * wave_total.
    #pragma unroll
    for (int off = 16; off >= 1; off >>= 1)
        s += __shfl_xor(s, off, 32);
    float wave_total = s * 0.0625f;  // exact /16

    const int lane = threadIdx.x & 31;
    const int wave = threadIdx.x >> 5;
    if (lane == 0) lds8[wave] = wave_total;
    __syncthreads();
    float tot = 0.0f;
    if (threadIdx.x == 0) {
        #pragma unroll
        for (int w = 0; w < BLOCK / 32; ++w) tot += lds8[w];
    }
    return tot;
}

// Pass 1: streaming grid-stride sweep. Each element i is loaded once as the
// "center"; the +1 (same cacheline ~75% of the time), +385 (1.5 KB ahead),
// and +385^2 (593 KB ahead, retained by the 192 MB L2) neighbors hit cache,
// keeping HBM traffic ~= 228 MB (the bandwidth floor).
__global__ void occ_conn_partials(const float* __restrict__ occ,
                                  float* __restrict__ partials) {
    const unsigned stride = blockDim.x * gridDim.x;
    float acc = 0.0f;
    for (unsigned i = blockIdx.x * blockDim.x + threadIdx.x; i < NTOT; i += stride) {
        const unsigned z = i % G;
        const unsigned t = i / G;
        const unsigned y = t % G;
        const unsigned x = t / G;
        const float c = occ[i];
        if (z < G - 1u) acc += fabsf(occ[i + 1u]    - c);
        if (y < G - 1u) acc += fabsf(occ[i + G]     - c);
        if (x < G - 1u) acc += fabsf(occ[i + PLANE] - c);
    }
    __shared__ float lds8[BLOCK / 32];
    const float tot = block_reduce_wmma(acc, lds8);
    if (threadIdx.x == 0) partials[blockIdx.x] = tot;
}

// Pass 2: single block, fixed-order tree -> bitwise-deterministic final sum.
__global__ void occ_conn_finalize(const float* __restrict__ partials,
                                  float* __restrict__ out, int n) {
    float acc = 0.0f;
    for (int i = threadIdx.x; i < n; i += BLOCK) acc += partials[i];
    __shared__ float lds8[BLOCK / 32];
    const float tot = block_reduce_wmma(acc, lds8);
    if (threadIdx.x == 0) out[0] = tot;
}

extern "C" void kernel_launch(void* const* d_in, const int* in_sizes, int n_in,
                              void* d_out, int out_size, void* d_ws, size_t ws_size,
                              hipStream_t stream) {
    const float* occ = (const float*)d_in[0];
    float* out = (float*)d_out;
    float* partials = (float*)d_ws;  // 4096 floats = 16 KB of scratch

    occ_conn_partials<<<NBLOCKS, BLOCK, 0, stream>>>(occ, partials);
    occ_conn_finalize<<<1, BLOCK, 0, stream>>>(partials, out, NBLOCKS);
}